// Encoder_1176821039646
// MI455X (gfx1250) — compile-verified
//
#include <hip/hip_runtime.h>
#include <math.h>

#define N_NODES 100000
#define N_EDGES 3200000
#define XDIM    2000
#define HID     128
#define NUM_CLS 20
#define NPB     8      // nodes per block in the node-update kernel

typedef __attribute__((ext_vector_type(2))) float v2f;
typedef __attribute__((ext_vector_type(8))) float v8f;

// ---------------------------------------------------------------------------
// Kernel 1: h = x @ W1 + b1   (fp32 WMMA 16x16x4)
// One wave owns a 16-row strip and computes all 128 output columns
// (8 accumulator tiles), so x is read from HBM exactly once.
//
// A-frag (16x4 f32, 2 VGPRs): lanes 0-15 -> M=lane, K={0,1}; lanes 16-31 -> K={2,3}
// B-frag (4x16 f32, 2 VGPRs): lanes 0-15 -> K={0,1}, N=lane; lanes 16-31 -> K={2,3}
// C/D    (16x16 f32, 8 VGPRs): vgpr r: lanes 0-15 -> (M=r, N=lane),
//                                       lanes 16-31 -> (M=r+8, N=lane-16)
// ---------------------------------------------------------------------------
__global__ __launch_bounds__(256) void gemm1_wmma(
    const float* __restrict__ x, const float* __restrict__ W1,
    const float* __restrict__ b1, float* __restrict__ feat)
{
  const int nstrip = N_NODES / 16;                       // 6250
  const int wave   = blockIdx.x * (blockDim.x >> 5) + (threadIdx.x >> 5);
  if (wave >= nstrip) return;                            // uniform per wave: EXEC stays full
  const int lane  = threadIdx.x & 31;
  const int lhalf = lane >> 4;                           // 0 or 1
  const int l16   = lane & 15;

  const int row = wave * 16 + l16;
  const float* ap = x + (size_t)row * XDIM + (lhalf << 1);   // 8B aligned

  v8f acc[8] = {};

  for (int k = 0; k < XDIM; k += 4) {
    v2f a = *(const v2f*)(ap + k);                       // global_load_b64
    const int   brow = k + (lhalf << 1);
    const float* bp  = W1 + (size_t)brow * HID + l16;
    #pragma unroll
    for (int t = 0; t < 8; ++t) {
      v2f b;
      b.x = bp[t * 16];
      b.y = bp[HID + t * 16];
      acc[t] = __builtin_amdgcn_wmma_f32_16x16x4_f32(
          /*neg_a=*/false, a, /*neg_b=*/false, b,
          /*c_mod=*/(short)0, acc[t], /*reuse_a=*/false, /*reuse_b=*/false);
    }
  }

  // epilogue: add bias, write pre-relu h into the 'feat' output region
  #pragma unroll
  for (int t = 0; t < 8; ++t) {
    const int   col  = t * 16 + l16;
    const float bias = b1[col];
    #pragma unroll
    for (int r = 0; r < 8; ++r) {
      const int orow = wave * 16 + r + lhalf * 8;
      feat[(size_t)orow * HID + col] = acc[t][r] + bias;
    }
  }
}

// ---------------------------------------------------------------------------
// Kernel 0: zero msg_sum + deg in workspace
// ---------------------------------------------------------------------------
__global__ void zero_f32(float* __restrict__ p, long long n)
{
  long long i = (long long)blockIdx.x * blockDim.x + threadIdx.x;
  if (i < n) p[i] = 0.0f;
}

// ---------------------------------------------------------------------------
// Kernel 2: edge scatter-add.  thread = (edge, feature); wave = 32 contiguous
// features of one edge -> coalesced gather + coalesced f32 atomics, all in L2.
// relu(h) is recomputed on the fly from the pre-relu 'feat' region.
// ---------------------------------------------------------------------------
__global__ __launch_bounds__(256) void scatter_edges(
    const long long* __restrict__ ei, const float* __restrict__ feat,
    float* __restrict__ msg, float* __restrict__ deg)
{
  long long idx = (long long)blockIdx.x * blockDim.x + threadIdx.x;
  long long e   = idx >> 7;
  int       f   = (int)(idx & 127);
  if (e >= (long long)N_EDGES) return;
  long long src = ei[e];                     // edge_index[0][e]
  long long dst = ei[(long long)N_EDGES + e];// edge_index[1][e]
  float v = fmaxf(feat[src * HID + f], 0.0f);
  atomicAdd(&msg[dst * HID + f], v);
  if (f == 0) atomicAdd(&deg[dst], 1.0f);
}

// ---------------------------------------------------------------------------
// Kernel 3: column-normalize Wn (128 x 20) once into workspace
// ---------------------------------------------------------------------------
__global__ void wn_colnorm(const float* __restrict__ Wn, float* __restrict__ WnN)
{
  int c = threadIdx.x;
  if (c >= NUM_CLS) return;
  float s = 0.0f;
  for (int k = 0; k < HID; ++k) { float w = Wn[k * NUM_CLS + c]; s += w * w; }
  float inv = 1.0f / fmaxf(sqrtf(s), 1e-12f);
  for (int k = 0; k < HID; ++k) WnN[k * NUM_CLS + c] = Wn[k * NUM_CLS + c] * inv;
}

// ---------------------------------------------------------------------------
// Kernel 4: per-node update.  128 threads handle NPB=8 nodes so each Wl/Wr
// element loaded from L2 is reused 8x from a register.
//   agg      = msg / max(deg,1)
//   out_feat = agg @ Wl + bl + relu(h) @ Wr
//   out      = 10 * rownorm(out_feat) @ WnN
// ---------------------------------------------------------------------------
__global__ __launch_bounds__(128) void node_kernel(
    const float* __restrict__ feat, const float* __restrict__ msg,
    const float* __restrict__ deg,  const float* __restrict__ Wl,
    const float* __restrict__ bl,   const float* __restrict__ Wr,
    const float* __restrict__ WnN,
    float* __restrict__ out, float* __restrict__ out_feat)
{
  __shared__ float agg_s[NPB][HID];
  __shared__ float hr_s[NPB][HID];
  __shared__ float outf_s[NPB][HID];
  __shared__ float scale_s[NPB];

  const int f  = threadIdx.x;          // 0..127
  const int n0 = blockIdx.x * NPB;

  #pragma unroll
  for (int j = 0; j < NPB; ++j) {
    const int node = n0 + j;
    const float d  = fmaxf(deg[node], 1.0f);
    agg_s[j][f] = msg[(size_t)node * HID + f] / d;
    hr_s[j][f]  = fmaxf(feat[(size_t)node * HID + f], 0.0f);
  }
  __syncthreads();

  float acc[NPB];
  const float biasv = bl[f];
  #pragma unroll
  for (int j = 0; j < NPB; ++j) acc[j] = biasv;

  for (int k = 0; k < HID; ++k) {
    const float wl = Wl[k * HID + f];
    const float wr = Wr[k * HID + f];
    #pragma unroll
    for (int j = 0; j < NPB; ++j)
      acc[j] += agg_s[j][k] * wl + hr_s[j][k] * wr;
  }

  #pragma unroll
  for (int j = 0; j < NPB; ++j) {
    outf_s[j][f] = acc[j];
    out_feat[(size_t)(n0 + j) * HID + f] = acc[j];
  }
  __syncthreads();

  if (f < NPB) {
    float s = 0.0f;
    for (int k = 0; k < HID; ++k) { float v = outf_s[f][k]; s += v * v; }
    scale_s[f] = 10.0f / fmaxf(sqrtf(s), 1e-12f);
  }
  __syncthreads();

  for (int idx = f; idx < NPB * NUM_CLS; idx += 128) {
    const int j = idx / NUM_CLS, c = idx % NUM_CLS;
    float s = 0.0f;
    for (int k = 0; k < HID; ++k) s += outf_s[j][k] * WnN[k * NUM_CLS + c];
    out[(size_t)(n0 + j) * NUM_CLS + c] = scale_s[j] * s;
  }
}

// ---------------------------------------------------------------------------
extern "C" void kernel_launch(void* const* d_in, const int* in_sizes, int n_in,
                              void* d_out, int out_size, void* d_ws, size_t ws_size,
                              hipStream_t stream)
{
  const float*     x  = (const float*)d_in[0];
  const long long* ei = (const long long*)d_in[1];   // int64 edge_index [2, E]
  const float*     W1 = (const float*)d_in[2];
  const float*     b1 = (const float*)d_in[3];
  const float*     Wl = (const float*)d_in[4];
  const float*     bl = (const float*)d_in[5];
  const float*     Wr = (const float*)d_in[6];
  const float*     Wn = (const float*)d_in[7];

  // outputs concatenated: out [N,20] | feat [N,128] | out_feat [N,128]
  float* out      = (float*)d_out;
  float* feat     = out + (size_t)N_NODES * NUM_CLS;
  float* out_feat = feat + (size_t)N_NODES * HID;

  // workspace: msg_sum [N,128] | deg [N] | WnN [128,20]
  float* msg = (float*)d_ws;
  float* deg = msg + (size_t)N_NODES * HID;
  float* WnN = deg + N_NODES;

  {
    long long n = (long long)N_NODES * HID + N_NODES;
    int blocks  = (int)((n + 255) / 256);
    zero_f32<<<blocks, 256, 0, stream>>>(msg, n);
  }
  {
    const int nstrip = N_NODES / 16;                 // 6250 waves
    const int wpb    = 8;                            // waves per 256-thread block
    int blocks = (nstrip + wpb - 1) / wpb;
    gemm1_wmma<<<blocks, 256, 0, stream>>>(x, W1, b1, feat);
  }
  wn_colnorm<<<1, 32, 0, stream>>>(Wn, WnN);
  {
    long long total = (long long)N_EDGES * HID;
    int blocks = (int)((total + 255) / 256);
    scatter_edges<<<blocks, 256, 0, stream>>>(ei, feat, msg, deg);
  }
  node_kernel<<<N_NODES / NPB, 128, 0, stream>>>(feat, msg, deg, Wl, bl, Wr,
                                                 WnN, out, out_feat);
}